// Final_910533067699
// MI455X (gfx1250) — compile-verified
//
#include <hip/hip_runtime.h>
#include <hip/hip_bf16.h>

typedef float v2f __attribute__((ext_vector_type(2)));
typedef float v8f __attribute__((ext_vector_type(8)));

#define B_  8
#define C_  64
#define N_  4096
#define KNN 20

// ---------------------------------------------------------------------------
// Kernel 0: squared norms  xx[b*N+n] = sum_c x[b,c,n]^2   (x is [B,C,N])
// ---------------------------------------------------------------------------
__global__ void sqnorm_kernel(const float* __restrict__ x, float* __restrict__ xx) {
  int t = blockIdx.x * blockDim.x + threadIdx.x;
  if (t >= B_ * N_) return;
  int b = t / N_, n = t % N_;
  const float* xb = x + (size_t)b * C_ * N_ + n;
  float s = 0.f;
  #pragma unroll
  for (int c = 0; c < C_; ++c) { float v = xb[(size_t)c * N_]; s = fmaf(v, v, s); }
  xx[t] = s;
}

// ---------------------------------------------------------------------------
// Kernel 1: fused Gram-matrix (f32 WMMA) + running top-K per row.
// One wave (32 lanes) owns a 16-row strip of one batch. A-panel (16x64) lives
// in 32 VGPRs for the whole strip. Column panels of 64 are computed as four
// 16x16 WMMA tiles (16 x K=4 steps each), converted to the top_k score
// s(i,j) = 2*G[i,j] - ||x_j||^2 (||x_i||^2 is constant per row, so ranking is
// identical to the reference's -||xi-xj||^2), staged through LDS, and folded
// into per-lane sorted top-20 lists (2 lanes per row -> merged at the end).
// ---------------------------------------------------------------------------
__launch_bounds__(32)
__global__ void knn_topk_kernel(const float* __restrict__ x,
                                const float* __restrict__ xx,
                                int* __restrict__ idx) {
  __shared__ float tile[16][68];        // 16 rows x 64 cols (+pad)
  __shared__ float lds_d[32][KNN];
  __shared__ int   lds_i[32][KNN];

  const int lane = threadIdx.x;         // 0..31
  const int b    = blockIdx.y;
  const int i0   = blockIdx.x * 16;
  const float* xb  = x  + (size_t)b * C_ * N_;
  const float* xxb = xx + (size_t)b * N_;

  const int nlo = lane & 15;
  const int hi  = lane >> 4;            // 0 or 1

  // --- A-panel: rows i0..i0+15, channels 0..63, as 16 K=4 fragments.
  // ISA layout (16x4 f32 A): lanes 0-15 -> K={0,1}, lanes 16-31 -> K={2,3}.
  v2f a[16];
  {
    const int khalf = hi * 2;
    #pragma unroll
    for (int kk = 0; kk < 16; ++kk) {
      int c0 = kk * 4 + khalf;
      a[kk].x = xb[(size_t)(c0    ) * N_ + i0 + nlo];
      a[kk].y = xb[(size_t)(c0 + 1) * N_ + i0 + nlo];
    }
  }

  // Per-lane running top-K (sorted descending), kept in VGPRs via full unroll.
  float bd[KNN]; int bi[KNN];
  #pragma unroll
  for (int q = 0; q < KNN; ++q) { bd[q] = -3.0e38f; bi[q] = 0; }

  for (int jt = 0; jt < N_; jt += 64) {
    if (jt + 64 < N_) __builtin_prefetch(xb + jt + 64, 0, 0);  // global_prefetch_b8

    #pragma unroll
    for (int s = 0; s < 4; ++s) {
      const int j0 = jt + s * 16;
      const float xxj = xxb[j0 + nlo];
      v8f acc = {};
      #pragma unroll
      for (int kk = 0; kk < 16; ++kk) {
        // B fragment (4x16): VGPR0 lanes0-15 = row K0, lanes16-31 = row K1;
        //                    VGPR1 lanes0-15 = row K2, lanes16-31 = row K3.
        int c0 = kk * 4 + hi;
        v2f bb;
        bb.x = xb[(size_t)(c0    ) * N_ + j0 + nlo];
        bb.y = xb[(size_t)(c0 + 2) * N_ + j0 + nlo];
        acc = __builtin_amdgcn_wmma_f32_16x16x4_f32(false, a[kk], false, bb,
                                                    (short)0, acc, false, false);
      }
      // D layout: VGPR r -> row (r + 8*hi), col nlo.
      #pragma unroll
      for (int r = 0; r < 8; ++r)
        tile[r + 8 * hi][s * 16 + nlo] = 2.0f * acc[r] - xxj;
    }
    __syncthreads();

    // Scan: lane owns row (lane&15), column half hi (32 candidates).
    {
      const int row = nlo, cb = hi * 32;
      for (int cc = 0; cc < 32; ++cc) {
        float d = tile[row][cb + cc];
        int   j = jt + cb + cc;
        if (d > bd[KNN - 1]) {
          float vd = d; int vi = j;
          #pragma unroll
          for (int q = 0; q < KNN; ++q) {
            bool g = vd > bd[q];
            float td = bd[q]; int ti = bi[q];
            if (g) { bd[q] = vd; bi[q] = vi; vd = td; vi = ti; }
          }
        }
      }
    }
    __syncthreads();
  }

  // Merge the two half-row lists (both sorted desc; ties favor lower cols = list0).
  #pragma unroll
  for (int q = 0; q < KNN; ++q) { lds_d[lane][q] = bd[q]; lds_i[lane][q] = bi[q]; }
  __syncthreads();
  if (lane < 16) {
    int p0 = 0, p1 = 0;
    int* op = idx + ((size_t)b * N_ + i0 + lane) * KNN;
    for (int q = 0; q < KNN; ++q) {
      float d0 = lds_d[lane][p0];
      float d1 = lds_d[lane + 16][p1];
      if (d0 >= d1) { op[q] = lds_i[lane][p0];      ++p0; }
      else          { op[q] = lds_i[lane + 16][p1]; ++p1; }
    }
  }
}

// ---------------------------------------------------------------------------
// Kernel 2: gather + concat(neighbors - center, center) -> [B, 2C, N, K].
// Thread = (b, n, k); per channel the K/n dims are contiguous -> coalesced
// stores. x is L2-resident, so the irregular neighbor reads stay on-chip.
// ---------------------------------------------------------------------------
__global__ void gather_kernel(const float* __restrict__ x,
                              const int* __restrict__ idx,
                              float* __restrict__ out) {
  int t = blockIdx.x * blockDim.x + threadIdx.x;
  if (t >= B_ * N_ * KNN) return;
  int k = t % KNN;
  int n = (t / KNN) % N_;
  int b = t / (KNN * N_);
  int j = idx[t];
  const float* xb = x + (size_t)b * C_ * N_;
  float* ob = out + (size_t)b * (2 * C_) * N_ * KNN + (size_t)n * KNN + k;
  const size_t cstride = (size_t)N_ * KNN;
  #pragma unroll 8
  for (int c = 0; c < C_; ++c) {
    float cen = xb[(size_t)c * N_ + n];
    float nb  = xb[(size_t)c * N_ + j];
    ob[(size_t)c        * cstride] = nb - cen;
    ob[(size_t)(c + C_) * cstride] = cen;
  }
}

// ---------------------------------------------------------------------------
extern "C" void kernel_launch(void* const* d_in, const int* in_sizes, int n_in,
                              void* d_out, int out_size, void* d_ws, size_t ws_size,
                              hipStream_t stream) {
  (void)in_sizes; (void)n_in; (void)out_size; (void)ws_size;
  const float* x = (const float*)d_in[0];      // [B, C, N] f32
  // d_in[1] is k (== 20), compiled in as KNN.

  // Workspace layout: xx (B*N f32 = 128 KB) then idx (B*N*KNN i32 = 2.62 MB).
  float* xx  = (float*)d_ws;
  int*   idx = (int*)((char*)d_ws + sizeof(float) * (size_t)B_ * N_);
  float* out = (float*)d_out;

  sqnorm_kernel<<<(B_ * N_ + 255) / 256, 256, 0, stream>>>(x, xx);

  dim3 grid(N_ / 16, B_);
  knn_topk_kernel<<<grid, 32, 0, stream>>>(x, xx, idx);

  int total = B_ * N_ * KNN;
  gather_kernel<<<(total + 255) / 256, 256, 0, stream>>>(x, idx, out);
}